// TuckerDecompositionModel_30296699306020
// MI455X (gfx1250) — compile-verified
//
#include <hip/hip_runtime.h>

typedef __attribute__((ext_vector_type(2))) float v2f;
typedef __attribute__((ext_vector_type(8))) float v8f;

// ---------------------------------------------------------------------------
// Kernel 1: permute core (a,b,c) -> bt (a,c,b) so that GEMM1's B operand has
// element (k=b, n=a*256+c) at bt[n*256 + b] (k-contiguous per column).
// LDS-tiled transpose of each 256x256 slab.
// ---------------------------------------------------------------------------
__global__ __launch_bounds__(256) void permute_core_kernel(
    const float* __restrict__ core, float* __restrict__ bt)
{
    __shared__ float tile[16][17];
    const int a  = blockIdx.z;
    const int b0 = blockIdx.y * 16;
    const int c0 = blockIdx.x * 16;
    const int tx = threadIdx.x, ty = threadIdx.y;
    tile[ty][tx] = core[((size_t)(a * 256 + b0 + ty)) * 256 + (c0 + tx)];
    __syncthreads();
    bt[((size_t)(a * 256 + c0 + ty)) * 256 + (b0 + tx)] = tile[tx][ty];
}

// ---------------------------------------------------------------------------
// f32 WMMA GEMM: D[M,N] = A[M,K] * B[K,N]
//   A row-major (lda);  B stored n-major/k-contiguous: element (k,n) at
//   B[n*ldb + k];  one wave computes a (MR*16) x (NR*16) tile with
//   V_WMMA_F32_16X16X4_F32.
// Fragment layout per ISA 7.12.2 (32-bit A 16x4): lane l<16 holds
// A[m0+l, k..k+1], lane l>=16 holds A[m0+l-16, k+2..k+3] -> one float2 load
// per lane. B mirrors A with M<->N. C/D: VGPR v -> row m0+v+8*(l/16),
// col n0+(l%16).
//
// NOTE on shape of this loop (empirically tuned against the gfx1250 backend):
//  - K is a RUNTIME parameter on purpose. With an unknown trip count the
//    scheduler unrolls by 4 and emits s_clause'd load batches with staggered
//    s_wait_loadcnt (1 full wait per 4 WMMAs). Every compile-time-K /
//    pragma-unroll / manual double-buffer variant tried produced either
//    serialized load->wait(0)->wmma chains or 32 v_mov accumulator copies
//    per iteration. The K%16 remainder path is dead code for K=256.
//  - NTN is compile-time so tile decomposition is shifts/mults, no division.
// ---------------------------------------------------------------------------
template <int MR, int NR, int NTN>
__global__ __launch_bounds__(128) void wmma_gemm_f32_kernel(
    const float* __restrict__ A, int lda,
    const float* __restrict__ B, int ldb,
    float* __restrict__ D, int ldd, int K)
{
    const int wave = blockIdx.x * (blockDim.x >> 5) + (threadIdx.x >> 5);
    const int lane = threadIdx.x & 31;
    const int l16  = lane & 15;
    const int half = lane >> 4;           // 0: K pair {0,1}, 1: K pair {2,3}

    const int m0 = (wave / NTN) * (MR * 16);
    const int n0 = (wave % NTN) * (NR * 16);

    v8f acc[MR][NR];
#pragma unroll
    for (int i = 0; i < MR; ++i)
#pragma unroll
        for (int j = 0; j < NR; ++j) {
            v8f z = {0.f, 0.f, 0.f, 0.f, 0.f, 0.f, 0.f, 0.f};
            acc[i][j] = z;
        }

    const float* aBase = A + (size_t)(m0 + l16) * lda + (half << 1);
    const float* bBase = B + (size_t)(n0 + l16) * ldb + (half << 1);

    for (int k = 0; k < K; k += 4) {
        v2f a[MR], b[NR];
#pragma unroll
        for (int i = 0; i < MR; ++i)
            a[i] = *(const v2f*)(aBase + (size_t)(i * 16) * lda + k);
#pragma unroll
        for (int j = 0; j < NR; ++j)
            b[j] = *(const v2f*)(bBase + (size_t)(j * 16) * ldb + k);
#pragma unroll
        for (int i = 0; i < MR; ++i)
#pragma unroll
            for (int j = 0; j < NR; ++j)
                acc[i][j] = __builtin_amdgcn_wmma_f32_16x16x4_f32(
                    /*neg_a=*/false, a[i], /*neg_b=*/false, b[j],
                    /*c_mod=*/(short)0, acc[i][j],
                    /*reuse_a=*/false, /*reuse_b=*/false);
    }

#pragma unroll
    for (int i = 0; i < MR; ++i) {
#pragma unroll
        for (int j = 0; j < NR; ++j) {
            float* dBase = D + (size_t)(m0 + i * 16 + half * 8) * ldd
                             + (n0 + j * 16 + l16);
#pragma unroll
            for (int v = 0; v < 8; ++v)
                dBase[(size_t)v * ldd] = acc[i][j][v];
        }
    }
}

// ---------------------------------------------------------------------------
// Kernel 3: per requested row q (id -> slab i = id/768, row j = id%768):
//   V[q,c] = sum_a f0[i,a] * T[j, a*256 + c]
// ---------------------------------------------------------------------------
__global__ __launch_bounds__(256) void gather_v_kernel(
    const int* __restrict__ ids, const float* __restrict__ f0,
    const float* __restrict__ T, float* __restrict__ V)
{
    const int q  = blockIdx.x;
    const int c  = threadIdx.x;
    const int id = ids[q];
    const int i  = id / 768;
    const int j  = id - i * 768;

    const float* trow  = T + (size_t)j * 10240 + c;  // columns a*256 + c
    const float* f0row = f0 + i * 40;

    float acc = 0.f;
#pragma unroll
    for (int a = 0; a < 40; ++a)
        acc = fmaf(f0row[a], trow[(size_t)a * 256], acc);

    V[(size_t)q * 256 + c] = acc;
}

// ---------------------------------------------------------------------------
// Orchestration.
// Inputs (setup_inputs order): x[8192*768] (unused), ids[8192] int32,
// core[40*256*256], f0[40*40], f1[768*256], f2[768*256]. Output: 8192x768 f32.
// Workspace layout: bt 10,485,760 B | T 31,457,280 B | V 8,388,608 B.
// ---------------------------------------------------------------------------
extern "C" void kernel_launch(void* const* d_in, const int* in_sizes, int n_in,
                              void* d_out, int out_size, void* d_ws, size_t ws_size,
                              hipStream_t stream)
{
    (void)in_sizes; (void)n_in; (void)out_size; (void)ws_size;

    const int*   ids  = (const int*)  d_in[1];
    const float* core = (const float*)d_in[2];
    const float* f0   = (const float*)d_in[3];
    const float* f1   = (const float*)d_in[4];
    const float* f2   = (const float*)d_in[5];
    float*       out  = (float*)d_out;

    char*  ws = (char*)d_ws;
    float* bt = (float*)(ws);                         // (a,c,b): 40*256*256
    float* T  = (float*)(ws + 10485760);              // (768, 10240)
    float* V  = (float*)(ws + 10485760 + 31457280);   // (8192, 256)

    // 1) core (a,b,c) -> bt (a,c,b)
    permute_core_kernel<<<dim3(16, 16, 40), dim3(16, 16, 1), 0, stream>>>(core, bt);

    // 2) T = f1 @ bt^T : M=768, N=10240, K=256. 64x16 wave tiles (A small, B big):
    //    tilesM = 12, tilesN = 640 -> 7680 waves -> 1920 blocks.
    wmma_gemm_f32_kernel<4, 1, 640><<<1920, 128, 0, stream>>>(
        f1, 256, bt, 256, T, 10240, 256);

    // 3) V[q] = f0[i_q] . T[j_q] slabs
    gather_v_kernel<<<8192, 256, 0, stream>>>(ids, f0, T, V);

    // 4) out = V @ f2^T : M=8192, N=768, K=256. 16x64 wave tiles (A big, B small):
    //    tilesM = 512, tilesN = 12 -> 6144 waves -> 1536 blocks.
    wmma_gemm_f32_kernel<1, 4, 12><<<1536, 128, 0, stream>>>(
        V, 256, f2, 256, out, 768, 256);
}